// KNNConvLayer_43104291783210
// MI455X (gfx1250) — compile-verified
//
#include <hip/hip_runtime.h>
#include <stdint.h>

// ---------------------------------------------------------------------------
// KNN (k=49 smallest of 32768 per output column) on MI455X / gfx1250.
//
// d2(i,j) via V_WMMA_F32_16X16X4_F32 with augmented vectors:
//   A'[i] = (-2x, -2y, |a|^2, 1),  B'[j] = (x, y, 1, |b|^2)
//   => A'[i] . B'[:,j] = |a|^2 + |b|^2 - 2 a.b = d^2.
//
// Selection on the monotonic key max_i32(bits(d2), bits(1e-12))
// (== bits(fmaxf(d2,1e-12)) for non-NaN d2):
//   sweep 1: 8-bit histogram -> per-column bucket containing the 49th key.
//   sweep 2 (fused): speculatively collect every (sqrt-bits,row) with
//     top8 <= bucket into a CAP buffer, while also building the level-1
//     histogram of the bucket for fallback.
//   If a column's candidate count fits CAP (typical: ~100-200 <= 256), the
//   buffer holds ALL keys <= the 49th with full 32-bit keys -> extraction is
//   exact.  Overflowing columns refine via the already-built histogram
//   (+passes 2-3 in pathological cases) and re-collect.  IEEE-exact sqrtf
//   runs only on collected candidates, so output distances match the
//   reference exactly, and (d_bits, idx) u64 packing reproduces lax.top_k's
//   ascending / low-index-tie order.  Distances are recomputed per sweep:
//   compute is nearly free and avoids the 1 GB distance-matrix HBM trip.
// ---------------------------------------------------------------------------

typedef float v2f __attribute__((ext_vector_type(2)));
typedef float v8f __attribute__((ext_vector_type(8)));

#define N_IN   32768
#define N_OUT  8192
#define KSEL   49
#define COLS   16        // output columns per workgroup
#define CAP    256       // candidate buffer entries per column (power of 2)
#define NTHREADS 256
#define NWAVES 8
#define NITER  (N_IN / (16 * NWAVES))   // 256 row-tile iterations per wave
#define D2_CLAMP_BITS 0x2b8cbccc        // bits of 1e-12f

// Build the augmented A'/B' arrays (row-major [n][4]) in workspace.
__global__ __launch_bounds__(256) void knn_prep(const float* __restrict__ sc,
                                                const float* __restrict__ oc,
                                                float* __restrict__ Aaug,
                                                float* __restrict__ Baug) {
  int t = blockIdx.x * blockDim.x + threadIdx.x;
  if (t < N_IN) {
    float x = sc[2 * t], y = sc[2 * t + 1];
    float n2 = x * x + y * y;
    float4 v = make_float4(-2.0f * x, -2.0f * y, n2, 1.0f);
    *reinterpret_cast<float4*>(Aaug + 4 * t) = v;
  } else if (t < N_IN + N_OUT) {
    int j = t - N_IN;
    float x = oc[2 * j], y = oc[2 * j + 1];
    float n2 = x * x + y * y;
    float4 v = make_float4(x, y, 1.0f, n2);
    *reinterpret_cast<float4*>(Baug + 4 * j) = v;
  }
}

// A-operand load (32-bit A 16x4 layout: lanes 0-15 hold K=0,1; lanes 16-31
// hold K=2,3; rows are 16B-aligned float4s -> one b64 load).
__device__ __forceinline__ v2f load_a(const float* __restrict__ Aaug,
                                      int i0, int l16, int half) {
  return *reinterpret_cast<const v2f*>(Aaug + 4 * (i0 + l16) + 2 * half);
}

__device__ __forceinline__ v8f tile_d2(v2f areg, v2f breg) {
  v8f c = {};
  return __builtin_amdgcn_wmma_f32_16x16x4_f32(
      /*neg_a=*/false, areg, /*neg_b=*/false, breg,
      /*c_mod=*/(short)0, c, /*reuse_a=*/false, /*reuse_b=*/false);
}

// key = bits(fmaxf(d2, 1e-12f)) for non-NaN d2, as one v_max_i32.
__device__ __forceinline__ unsigned int d2_key(float d2) {
  int ib = (int)__float_as_uint(d2);
  return (unsigned int)(ib < (int)D2_CLAMP_BITS ? (int)D2_CLAMP_BITS : ib);
}

// Software-pipelined full-row sweep: the next A operand loads before the
// current tile's WMMA consumes its operand.
template <typename F>
__device__ __forceinline__ void sweep(const float* __restrict__ Aaug,
                                      int w, int l16, int half, F&& f) {
  v2f acur = load_a(Aaug, w * 16, l16, half);
#pragma unroll 4
  for (int it = 0; it < NITER - 1; ++it) {
    v2f anext = load_a(Aaug, ((it + 1) * NWAVES + w) * 16, l16, half);
    f(acur, (it * NWAVES + w) * 16);
    acur = anext;
  }
  f(acur, ((NITER - 1) * NWAVES + w) * 16);
}

__global__ __launch_bounds__(256) void knn_select(const float* __restrict__ Aaug,
                                                  const float* __restrict__ Baug,
                                                  int* __restrict__ outIdx,
                                                  float* __restrict__ outDist) {
  __shared__ unsigned int       s_hist[COLS * 256];   // 16 KB
  __shared__ unsigned long long s_buf [COLS * CAP];   // 32 KB
  __shared__ unsigned int       s_prefix[COLS];
  __shared__ unsigned int       s_need[COLS];
  __shared__ unsigned int       s_shift[COLS];
  __shared__ unsigned int       s_res[COLS];
  __shared__ unsigned int       s_fail[COLS];
  __shared__ unsigned int       s_cnt[COLS];
  __shared__ int                s_nact;

  const int tid  = threadIdx.x;
  const int w    = tid >> 5;       // wave id 0..7
  const int lane = tid & 31;
  const int half = lane >> 4;      // which K-pair this lane supplies
  const int l16  = lane & 15;      // this lane's column within the tile
  const int j0   = blockIdx.x * COLS;

  // B' tile lives in VGPRs for the whole kernel.
  const v2f breg = *reinterpret_cast<const v2f*>(Baug + 4 * (j0 + l16) + 2 * half);

  // Scan helper: wave w resolves columns 2w, 2w+1 from s_hist.
  auto do_scan = [&](int shiftNew, bool checkRes, bool forceRes) {
#pragma unroll
    for (int cc = 0; cc < 2; ++cc) {
      const int col = w * 2 + cc;
      if (checkRes && s_res[col]) continue;   // uniform per wave
      const unsigned int need = s_need[col];
      unsigned int s = 0;
#pragma unroll
      for (int b = 0; b < 8; ++b) s += s_hist[col * 256 + lane * 8 + b];
      unsigned int cum = s;
#pragma unroll
      for (int off = 1; off < 32; off <<= 1) {
        unsigned int v = __shfl_up(cum, off);
        if (lane >= off) cum += v;
      }
      unsigned int mask = (unsigned int)__ballot(cum >= need);
      int firstLane = __ffs(mask) - 1;
      if (lane == firstLane) {
        unsigned int excl = cum - s;
#pragma unroll
        for (int b = 0; b < 8; ++b) {
          unsigned int cb = s_hist[col * 256 + lane * 8 + b];
          if (excl + cb >= need) {
            unsigned int newNeed = need - excl;
            s_need[col]   = newNeed;
            s_prefix[col] = (s_prefix[col] << 8) | (unsigned int)(lane * 8 + b);
            s_shift[col]  = (unsigned int)shiftNew;
            if (checkRes &&
                ((((unsigned int)KSEL - newNeed) + cb <= CAP) || forceRes))
              s_res[col] = 1u;
            break;
          }
          excl += cb;
        }
      }
    }
  };

  if (tid < COLS) { s_prefix[tid] = 0u; s_need[tid] = KSEL; }

  // ---- Sweep 1: branchless level-0 histogram (bin = top 8 key bits) ----
  __syncthreads();
  for (int i = tid; i < COLS * 256; i += NTHREADS) s_hist[i] = 0u;
  __syncthreads();
  sweep(Aaug, w, l16, half, [&](v2f areg, int) {
    v8f c = tile_d2(areg, breg);
#pragma unroll
    for (int r = 0; r < 8; ++r) {
      unsigned int key = d2_key(c[r]);
      atomicAdd(&s_hist[(unsigned)l16 * 256u + (key >> 24)], 1u);
    }
  });
  __syncthreads();
  do_scan(24, false, false);
  __syncthreads();

  // ---- Sweep 2 (fused): speculative collect at top-8 prefix + level-1
  //      histogram of the chosen bucket (for fallback refinement). ----
  for (int i = tid; i < COLS * 256; i += NTHREADS) s_hist[i] = 0u;
  for (int i = tid; i < COLS * CAP; i += NTHREADS) s_buf[i] = ~0ull;
  if (tid < COLS) s_cnt[tid] = 0u;
  const unsigned int pref8 = s_prefix[l16];
  __syncthreads();
  sweep(Aaug, w, l16, half, [&](v2f areg, int i0) {
    v8f c = tile_d2(areg, breg);
#pragma unroll
    for (int r = 0; r < 8; ++r) {
      unsigned int key = d2_key(c[r]);
      unsigned int top8 = key >> 24;
      if (top8 <= pref8) {                       // rare (~0.6%)
        if (top8 == pref8)
          atomicAdd(&s_hist[(unsigned)l16 * 256u + ((key >> 16) & 255u)], 1u);
        float d = sqrtf(__uint_as_float(key));   // IEEE-exact, winners only
        unsigned int pos = atomicAdd(&s_cnt[l16], 1u);
        if (pos < CAP) {
          int row = i0 + r + (half << 3);
          s_buf[(unsigned)l16 * CAP + pos] =
              ((unsigned long long)__float_as_uint(d) << 32) | (unsigned int)row;
        }
      }
    }
  });
  __syncthreads();

  // ---- Overflow check; fallback refinement + re-collect if needed ----
  if (tid < COLS) {
    unsigned int failed = (s_cnt[tid] > CAP) ? 1u : 0u;
    s_fail[tid] = failed;
    s_res[tid]  = failed ? 0u : 1u;   // failed columns are "unresolved"
  }
  __syncthreads();
  if (tid == 0) {
    int n = 0;
#pragma unroll
    for (int c = 0; c < COLS; ++c) n += (int)s_fail[c];
    s_nact = n;
  }
  __syncthreads();

  if (s_nact > 0) {                    // uniform; rare path
    // Refine failed columns with the level-1 histogram already built.
    do_scan(16, true, false);
    __syncthreads();
    if (tid == 0) {
      int n = 0;
#pragma unroll
      for (int c = 0; c < COLS; ++c) n += (s_res[c] == 0u) ? 1 : 0;
      s_nact = n;
    }
    // Passes 2-3 for still-unresolved columns (pathological duplicates).
    for (int pass = 2; pass < 4; ++pass) {
      __syncthreads();
      if (s_nact == 0) break;
      for (int i = tid; i < COLS * 256; i += NTHREADS) s_hist[i] = 0u;
      __syncthreads();
      {
        const int shiftHi = 32 - 8 * pass;
        const int shift   = 24 - 8 * pass;
        const unsigned int pref = s_prefix[l16];
        const unsigned int resv = s_res[l16];
        sweep(Aaug, w, l16, half, [&](v2f areg, int) {
          v8f c = tile_d2(areg, breg);
#pragma unroll
          for (int r = 0; r < 8; ++r) {
            unsigned int key = d2_key(c[r]);
            if (!resv && ((key >> shiftHi) == pref))
              atomicAdd(&s_hist[(unsigned)l16 * 256u + ((key >> shift) & 255u)], 1u);
          }
        });
      }
      __syncthreads();
      do_scan(24 - 8 * pass, true, pass == 3);
      __syncthreads();
      if (tid == 0) {
        int n = 0;
#pragma unroll
        for (int c = 0; c < COLS; ++c) n += (s_res[c] == 0u) ? 1 : 0;
        s_nact = n;
      }
    }
    __syncthreads();
    // Reset and re-collect only the failed columns at their refined prefix.
    for (int i = tid; i < COLS * CAP; i += NTHREADS)
      if (s_fail[i >> 8]) s_buf[i] = ~0ull;
    if (tid < COLS && s_fail[tid]) s_cnt[tid] = 0u;
    __syncthreads();
    {
      const unsigned int failv = s_fail[l16];
      const unsigned int pref  = s_prefix[l16];
      const unsigned int sh    = s_shift[l16];
      sweep(Aaug, w, l16, half, [&](v2f areg, int i0) {
        v8f c = tile_d2(areg, breg);
#pragma unroll
        for (int r = 0; r < 8; ++r) {
          unsigned int key = d2_key(c[r]);
          if (failv && ((key >> sh) <= pref)) {
            float d = sqrtf(__uint_as_float(key));
            unsigned int pos = atomicAdd(&s_cnt[l16], 1u);
            if (pos < CAP) {
              int row = i0 + r + (half << 3);
              s_buf[(unsigned)l16 * CAP + pos] =
                  ((unsigned long long)__float_as_uint(d) << 32) | (unsigned int)row;
            }
          }
        }
      });
    }
  }
  __syncthreads();

  // ---- Extraction: 49 sorted minima per column; (d_bits,idx) packed u64
  //      => ascending distance, ties toward lower index (lax.top_k order).
  //      Only the live prefix of the buffer (s_cnt entries) is scanned. ----
  for (int cc = 0; cc < 2; ++cc) {
    const int col = w * 2 + cc;
    volatile unsigned long long* vb = &s_buf[col * CAP];
    unsigned int n = s_cnt[col];
    if (n > CAP) n = CAP;
    int chunks = (int)((n + 31u) >> 5);
    if (chunks < 1) chunks = 1;
    for (int r = 0; r < KSEL; ++r) {
      unsigned long long best = ~0ull;
      int bestPos = 0;
      for (int t = 0; t < chunks; ++t) {
        int pos = lane + t * 32;
        unsigned long long v = vb[pos];
        if (v < best) { best = v; bestPos = pos; }
      }
      unsigned long long gmin = best;
#pragma unroll
      for (int off = 16; off > 0; off >>= 1) {
        unsigned long long o = __shfl_xor(gmin, off);
        gmin = (o < gmin) ? o : gmin;
      }
      unsigned int m = (unsigned int)__ballot(best == gmin);
      int winner = __ffs(m) - 1;
      if (lane == winner) {
        vb[bestPos] = ~0ull;  // remove; LDS is in-order within a wave
        int j = j0 + col;
        outIdx[r * N_OUT + j]  = (int)(unsigned int)(gmin & 0xffffffffull);
        outDist[r * N_OUT + j] = __uint_as_float((unsigned int)(gmin >> 32));
      }
    }
  }
}

extern "C" void kernel_launch(void* const* d_in, const int* in_sizes, int n_in,
                              void* d_out, int out_size, void* d_ws, size_t ws_size,
                              hipStream_t stream) {
  (void)in_sizes; (void)n_in; (void)out_size; (void)ws_size;
  const float* sc = (const float*)d_in[0];   // sample_coords [32768,2] f32
  const float* oc = (const float*)d_in[1];   // out_coords   [8192,2]  f32
  float* Aaug = (float*)d_ws;                       // [32768,4] = 512 KB
  float* Baug = Aaug + (size_t)N_IN * 4;            // [8192,4]  = 128 KB
  int*   outIdx  = (int*)d_out;                     // [49,8192] int32
  float* outDist = (float*)d_out + (size_t)KSEL * N_OUT;  // [49,8192] f32

  knn_prep<<<(N_IN + N_OUT + 255) / 256, 256, 0, stream>>>(sc, oc, Aaug, Baug);
  knn_select<<<N_OUT / COLS, NTHREADS, 0, stream>>>(Aaug, Baug, outIdx, outDist);
}